// simple_attn_3d_19413252178675
// MI455X (gfx1250) — compile-verified
//
#include <hip/hip_runtime.h>
#include <math.h>

// ---------------------------------------------------------------------------
// Linear-attention 3D block for MI455X (gfx1250), fp32 via V_WMMA_F32_16X16X4.
// B=2, C=128, D=16, H=W=64  ->  N = 131072 positions, HW = 4096, BD = 32.
// Round 2: fragment reuse (pair/quad tiles per wave) + software-pipelined
// k-loops so WMMAs overlap outstanding global/LDS fragment loads.
// ---------------------------------------------------------------------------

#define CCH   128          // channels
#define HEADS 4
#define HC    32           // head channels
#define HW    4096
#define DHW   65536        // D*H*W
#define BDN   32           // B*D
#define PITCH 34           // LDS row pitch (even -> float2-aligned; conflict-lite)

typedef float v2f __attribute__((ext_vector_type(2)));
typedef float v8f __attribute__((ext_vector_type(8)));

static __device__ inline v8f wmma4(v2f a, v2f b, v8f c) {
  // D(16x16 f32) = A(16x4 f32) * B(4x16 f32) + C
  return __builtin_amdgcn_wmma_f32_16x16x4_f32(
      /*neg_a=*/false, a, /*neg_b=*/false, b,
      /*c_mod=*/(short)0, c, /*reuse_a=*/false, /*reuse_b=*/false);
}

// ---- Pair GEMM, K=128: one A row-fragment stream (global), two B column
//      tiles (LDS cols nl and nl+16). Software pipelined. -------------------
static __device__ inline void gemm_pair_k128(const float* __restrict__ arow,
                                             const float* __restrict__ Bb,
                                             v8f& c0, v8f& c1) {
  v2f a  = *(const v2f*)(arow);
  v2f b0 = { Bb[0],  Bb[PITCH] };
  v2f b1 = { Bb[16], Bb[PITCH + 16] };
  for (int k0 = 0; k0 < 124; k0 += 4) {
    v2f an = *(const v2f*)(arow + k0 + 4);
    const float* Bn = Bb + (k0 + 4) * PITCH;
    v2f b0n = { Bn[0],  Bn[PITCH] };
    v2f b1n = { Bn[16], Bn[PITCH + 16] };
    c0 = wmma4(a, b0, c0);
    c1 = wmma4(a, b1, c1);
    a = an; b0 = b0n; b1 = b1n;
  }
  c0 = wmma4(a, b0, c0);
  c1 = wmma4(a, b1, c1);
}

// ---- Quad GEMM, K=128: two A row-fragment streams, two B column tiles.
//      2 A loads + 2 B loads feed 4 WMMAs per k-step. -----------------------
static __device__ inline void gemm_quad_k128(const float* __restrict__ arow0,
                                             const float* __restrict__ arow1,
                                             const float* __restrict__ Bb,
                                             v8f& c00, v8f& c01,
                                             v8f& c10, v8f& c11) {
  v2f a0 = *(const v2f*)(arow0);
  v2f a1 = *(const v2f*)(arow1);
  v2f b0 = { Bb[0],  Bb[PITCH] };
  v2f b1 = { Bb[16], Bb[PITCH + 16] };
  for (int k0 = 0; k0 < 124; k0 += 4) {
    v2f a0n = *(const v2f*)(arow0 + k0 + 4);
    v2f a1n = *(const v2f*)(arow1 + k0 + 4);
    const float* Bn = Bb + (k0 + 4) * PITCH;
    v2f b0n = { Bn[0],  Bn[PITCH] };
    v2f b1n = { Bn[16], Bn[PITCH + 16] };
    c00 = wmma4(a0, b0, c00);
    c01 = wmma4(a0, b1, c01);
    c10 = wmma4(a1, b0, c10);
    c11 = wmma4(a1, b1, c11);
    a0 = a0n; a1 = a1n; b0 = b0n; b1 = b1n;
  }
  c00 = wmma4(a0, b0, c00);
  c01 = wmma4(a0, b1, c01);
  c10 = wmma4(a1, b0, c10);
  c11 = wmma4(a1, b1, c11);
}

// ---- Pair GEMM, K=32, A and both B from LDS (row-contiguous b64 frags) ----
static __device__ inline void kv_pair_k32(const float* __restrict__ Ar,
                                          const float* __restrict__ Br0,
                                          const float* __restrict__ Br1,
                                          v8f& c0, v8f& c1) {
  v2f a  = *(const v2f*)(Ar);
  v2f b0 = *(const v2f*)(Br0);
  v2f b1 = *(const v2f*)(Br1);
  for (int k0 = 0; k0 < 28; k0 += 4) {
    v2f an  = *(const v2f*)(Ar  + k0 + 4);
    v2f b0n = *(const v2f*)(Br0 + k0 + 4);
    v2f b1n = *(const v2f*)(Br1 + k0 + 4);
    c0 = wmma4(a, b0, c0);
    c1 = wmma4(a, b1, c1);
    a = an; b0 = b0n; b1 = b1n;
  }
  c0 = wmma4(a, b0, c0);
  c1 = wmma4(a, b1, c1);
}

// ---------------------------------------------------------------------------
// K1: project k,v (WMMA, K=128), LayerNorm (ddof=1, /(std+eps)), accumulate
//     per-head k^T v (WMMA over positions) -> atomicAdd into kvraw[bd][h][c][d].
// Grid: 32 bd * 32 strips = 1024 blocks, 256 threads. Strip = 128 pos in 4
// chunks of 32; kv accumulators persist in VGPRs across chunks.
// ---------------------------------------------------------------------------
__global__ __launch_bounds__(256) void k1_kv_accum(
    const float* __restrict__ x,
    const float* __restrict__ qkv_w, const float* __restrict__ qkv_b,
    const float* __restrict__ kln_w, const float* __restrict__ kln_b,
    const float* __restrict__ vln_w, const float* __restrict__ vln_b,
    float* __restrict__ kvraw) {
  __shared__ float xs[CCH * PITCH];
  __shared__ float kb[CCH * PITCH];
  __shared__ float vb[CCH * PITCH];

  const int tid  = threadIdx.x;
  const int lane = tid & 31, wave = tid >> 5;
  const int nl   = lane & 15, hi = lane >> 4;

  const int bd    = blockIdx.x >> 5;   // 32 strips per bd
  const int strip = blockIdx.x & 31;
  const int b     = bd >> 4, d = bd & 15;
  const int xbase = b * (CCH * DHW) + d * HW + strip * 128;

  // ---- projection quad for this wave: mat (k/v), m-tiles m0a/m0b, n 0..31 ----
  const int mat = wave >> 2;           // waves 0-3: k, waves 4-7: v
  const int w4  = wave & 3;
  const int m0a = w4 * 32, m0b = m0a + 16;
  const int ra  = m0a + nl, rb = m0b + nl;
  const float* arow0 = qkv_w + ((ra >> 5) * 96 + 32 + mat * 32 + (ra & 31)) * CCH + 2 * hi;
  const float* arow1 = qkv_w + ((rb >> 5) * 96 + 32 + mat * 32 + (rb & 31)) * CCH + 2 * hi;
  float* projdst = mat ? vb : kb;

  // ---- kv accumulation pair: head h, m-tile mt, n-tiles 0 and 1 ----
  const int h  = wave >> 1;            // t0=2w, t1=2w+1 share h and mt
  const int mt = wave & 1;
  v8f acc0 = {}; v8f acc1 = {};

  for (int chunk = 0; chunk < 4; ++chunk) {
    const int hwoff = chunk * 32;
    __syncthreads();                                   // protect LDS reuse
    // ---- load x chunk [128 ch][32 pos], coalesced over pos ----
    for (int i = 0; i < 16; ++i) {
      int idx = i * 256 + tid;
      int c = idx >> 5, p = idx & 31;
      xs[c * PITCH + p] = x[xbase + c * DHW + hwoff + p];
    }
    __syncthreads();
    // ---- k/v projection: 2x2 quad per wave, shared fragments ----
    {
      v8f c00 = {}, c01 = {}, c10 = {}, c11 = {};
      gemm_quad_k128(arow0, arow1, xs + 2 * hi * PITCH + nl, c00, c01, c10, c11);
      for (int r = 0; r < 8; ++r) {
        int ma = m0a + r + 8 * hi, mb = m0b + r + 8 * hi;
        float ba = qkv_b[(ma >> 5) * 96 + 32 + mat * 32 + (ma & 31)];
        float bb = qkv_b[(mb >> 5) * 96 + 32 + mat * 32 + (mb & 31)];
        projdst[ma * PITCH + nl]      = c00[r] + ba;
        projdst[ma * PITCH + 16 + nl] = c01[r] + ba;
        projdst[mb * PITCH + nl]      = c10[r] + bb;
        projdst[mb * PITCH + 16 + nl] = c11[r] + bb;
      }
    }
    __syncthreads();
    // ---- LayerNorm over headc=32 (unbiased var, eps on std) ----
    {
      int lmat = tid >> 7;             // threads 0-127: k, 128-255: v
      int t    = tid & 127;
      int lh   = t >> 5, p = t & 31;
      float* buf = lmat ? vb : kb;
      const float* lw = lmat ? vln_w : kln_w;
      const float* lb = lmat ? vln_b : kln_b;
      float s = 0.f, s2 = 0.f;
      for (int e = 0; e < 32; ++e) {
        float v = buf[(lh * 32 + e) * PITCH + p];
        s += v; s2 += v * v;
      }
      float mean = s * (1.0f / 32.0f);
      float var  = fmaxf((s2 - 32.0f * mean * mean) * (1.0f / 31.0f), 0.0f);
      float inv  = 1.0f / (sqrtf(var) + 1e-5f);
      for (int e = 0; e < 32; ++e) {
        float v = buf[(lh * 32 + e) * PITCH + p];
        buf[(lh * 32 + e) * PITCH + p] =
            lw[lh * 32 + e] * ((v - mean) * inv) + lb[lh * 32 + e];
      }
    }
    __syncthreads();
    // ---- kv += k^T v over 32 positions (shared A fragment) ----
    kv_pair_k32(kb + (h * 32 + mt * 16 + nl) * PITCH + 2 * hi,
                vb + (h * 32 + nl)           * PITCH + 2 * hi,
                vb + (h * 32 + 16 + nl)      * PITCH + 2 * hi,
                acc0, acc1);
  }
  // ---- commit partial kv sums ----
  float* base = kvraw + ((bd * 4 + h) * 32 + mt * 16) * 32;
  for (int r = 0; r < 8; ++r) {
    atomicAdd(base + (r + 8 * hi) * 32 + nl,      acc0[r]);
    atomicAdd(base + (r + 8 * hi) * 32 + 16 + nl, acc1[r]);
  }
}

// ---------------------------------------------------------------------------
// K2: Aw[bd][o][c] = sum_e kv[h][e][d] * qkv_w[h*96+e][c] / HW  (o = h*32+d)
//     Ab[bd][o]    = sum_e kv[h][e][d] * qkv_b[h*96+e]          (q-bias fold)
// ---------------------------------------------------------------------------
__global__ __launch_bounds__(256) void k2_make_A(
    const float* __restrict__ kvraw,
    const float* __restrict__ qkv_w, const float* __restrict__ qkv_b,
    float* __restrict__ Aw, float* __restrict__ Ab) {
  __shared__ float kv[HEADS * HC * HC];
  const int bd = blockIdx.x, tid = threadIdx.x;
  for (int i = tid; i < HEADS * HC * HC; i += 256)
    kv[i] = kvraw[bd * HEADS * HC * HC + i] * (1.0f / (float)HW);
  __syncthreads();
  for (int idx = tid; idx < CCH * CCH; idx += 256) {
    int o = idx >> 7, c = idx & 127;
    int h = o >> 5, dd = o & 31;
    float s = 0.f;
    for (int e = 0; e < 32; ++e)
      s += kv[(h * 32 + e) * 32 + dd] * qkv_w[(h * 96 + e) * CCH + c];
    Aw[bd * CCH * CCH + idx] = s;
  }
  if (tid < CCH) {
    int o = tid, h = o >> 5, dd = o & 31;
    float s = 0.f;
    for (int e = 0; e < 32; ++e)
      s += kv[(h * 32 + e) * 32 + dd] * qkv_b[h * 96 + e];
    Ab[bd * CCH + o] = s;
  }
}

// ---------------------------------------------------------------------------
// K3: fully fused tail, per 32-position tile:
//   ret = Aw_bd @ x + Ab + x ;  h = gelu(o1 @ ret + o1b) ;
//   out = o2 @ h + o2b + x     (exact erf GELU, all WMMA fp32, K=128)
// Grid: 4096 blocks, 256 threads; wave w owns row-tile m0=w*16, cols 0..31.
// ---------------------------------------------------------------------------
__global__ __launch_bounds__(256) void k3_fused_mlp(
    const float* __restrict__ x,
    const float* __restrict__ Aw, const float* __restrict__ Ab,
    const float* __restrict__ o1_w, const float* __restrict__ o1_b,
    const float* __restrict__ o2_w, const float* __restrict__ o2_b,
    float* __restrict__ out) {
  __shared__ float xs[CCH * PITCH];
  __shared__ float rs[CCH * PITCH];
  __shared__ float hs[CCH * PITCH];

  const int tid  = threadIdx.x;
  const int lane = tid & 31, wave = tid >> 5;
  const int nl   = lane & 15, hi = lane >> 4;

  const int bd   = blockIdx.x >> 7;                    // 128 tiles per bd
  const int tile = blockIdx.x & 127;
  const int b    = bd >> 4, d = bd & 15;
  const int xbase = b * (CCH * DHW) + d * HW + tile * 32;

  for (int i = 0; i < 16; ++i) {
    int idx = i * 256 + tid;
    int c = idx >> 5, p = idx & 31;
    xs[c * PITCH + p] = x[xbase + c * DHW + p];
  }
  __syncthreads();

  const int m0 = wave * 16;
  const int arow_off = (m0 + nl) * CCH + 2 * hi;
  const int boff = 2 * hi * PITCH + nl;

  // ---- stage 1: rs = Aw @ xs + Ab + xs (residual) ----
  {
    v8f c0 = {}, c1 = {};
    gemm_pair_k128(Aw + bd * CCH * CCH + arow_off, xs + boff, c0, c1);
    const float* AbB = Ab + bd * CCH;
    for (int r = 0; r < 8; ++r) {
      int m = m0 + r + 8 * hi;
      float bias = AbB[m];
      rs[m * PITCH + nl]      = c0[r] + bias + xs[m * PITCH + nl];
      rs[m * PITCH + 16 + nl] = c1[r] + bias + xs[m * PITCH + 16 + nl];
    }
  }
  __syncthreads();

  // ---- stage 2: hs = gelu(o1 @ rs + o1_b) ----
  {
    v8f c0 = {}, c1 = {};
    gemm_pair_k128(o1_w + arow_off, rs + boff, c0, c1);
    for (int r = 0; r < 8; ++r) {
      int m = m0 + r + 8 * hi;
      float v0 = c0[r] + o1_b[m];
      float v1 = c1[r] + o1_b[m];
      hs[m * PITCH + nl]      = 0.5f * v0 * (1.0f + erff(v0 * 0.70710678118654752f));
      hs[m * PITCH + 16 + nl] = 0.5f * v1 * (1.0f + erff(v1 * 0.70710678118654752f));
    }
  }
  __syncthreads();

  // ---- stage 3: out = o2 @ hs + o2_b + xs -> global ----
  {
    v8f c0 = {}, c1 = {};
    gemm_pair_k128(o2_w + arow_off, hs + boff, c0, c1);
    for (int r = 0; r < 8; ++r) {
      int m = m0 + r + 8 * hi;
      float bias = o2_b[m];
      out[xbase + m * DHW + nl]      = c0[r] + bias + xs[m * PITCH + nl];
      out[xbase + m * DHW + 16 + nl] = c1[r] + bias + xs[m * PITCH + 16 + nl];
    }
  }
}

// ---------------------------------------------------------------------------
extern "C" void kernel_launch(void* const* d_in, const int* in_sizes, int n_in,
                              void* d_out, int out_size, void* d_ws, size_t ws_size,
                              hipStream_t stream) {
  const float* x     = (const float*)d_in[0];
  const float* qkv_w = (const float*)d_in[1];
  const float* qkv_b = (const float*)d_in[2];
  const float* o1_w  = (const float*)d_in[3];
  const float* o1_b  = (const float*)d_in[4];
  const float* o2_w  = (const float*)d_in[5];
  const float* o2_b  = (const float*)d_in[6];
  const float* kln_w = (const float*)d_in[7];
  const float* kln_b = (const float*)d_in[8];
  const float* vln_w = (const float*)d_in[9];
  const float* vln_b = (const float*)d_in[10];
  float* out = (float*)d_out;

  // workspace: kvraw (512KB) | Aw (2MB) | Ab (16KB)  -> ~2.6 MB total
  float* kvraw = (float*)d_ws;
  float* Aw    = kvraw + BDN * HEADS * HC * HC;        // +131072
  float* Ab    = Aw + BDN * CCH * CCH;                 // +524288

  hipMemsetAsync(kvraw, 0, (size_t)BDN * HEADS * HC * HC * sizeof(float), stream);
  k1_kv_accum<<<BDN * 32, 256, 0, stream>>>(x, qkv_w, qkv_b, kln_w, kln_b,
                                            vln_w, vln_b, kvraw);
  k2_make_A<<<BDN, 256, 0, stream>>>(kvraw, qkv_w, qkv_b, Aw, Ab);
  k3_fused_mlp<<<BDN * 128, 256, 0, stream>>>(x, Aw, Ab, o1_w, o1_b,
                                              o2_w, o2_b, out);
}